// MHAttention_22557168239313
// MI455X (gfx1250) — compile-verified
//
#include <hip/hip_runtime.h>
#include <hip/hip_bf16.h>

// MHA forward for MI455X (gfx1250, wave32, WMMA, async-LDS copies).
// Pipeline:
//   1) gemm_kernel x3 : Q/K/V projections (WMMA f32, async double-buffered
//                       LDS staging), epilogue scatters into head layout
//   2) normalize_rows : L2-normalize Qh, Kh rows (HD=64)
//   3) attn_stats     : WMMA logits, per-wave async K-tile staging, online
//                       per-row softmax stats (max / sum)
//   4) attn_out       : WMMA logits recompute, write normalized attn (537MB,
//                       the dominant HBM traffic), WMMA-accumulate z = p @ V
//                       with per-wave async K+V tile staging
//   5) gemm_kernel    : zc @ Wz + bz -> d_out

typedef __attribute__((ext_vector_type(2))) float v2f;
typedef __attribute__((ext_vector_type(8))) float v8f;

#define B_   2
#define S_   2048
#define E_   1024
#define H_   16
#define HD_  64
#define BH_  (B_ * H_)
#define BSE_ (B_ * S_ * E_)   // 4194304 elements

static __device__ __forceinline__ v8f wmma_f32(v2f a, v2f b, v8f c) {
  // D = A(16x4) * B(4x16) + C(16x16), fp32
  return __builtin_amdgcn_wmma_f32_16x16x4_f32(
      /*neg_a=*/false, a, /*neg_b=*/false, b,
      /*c_mod=*/(short)0, c, /*reuse_a=*/false, /*reuse_b=*/false);
}

// Async global->LDS copy, 16B per lane. ASYNCcnt-tracked (wave-local).
// LDS operand is the low 32 bits of the flat shared-pointer (== LDS offset).
static __device__ __forceinline__ void async_copy_b128(const float* gsrc,
                                                       const float* ldst) {
  asm volatile("global_load_async_to_lds_b128 %0, %1, off"
               :: "v"((unsigned)(size_t)ldst),
                  "v"((unsigned long long)(size_t)gsrc)
               : "memory");
}
#define WAIT_ASYNC(n) asm volatile("s_wait_asynccnt " #n ::: "memory")

// ---------------------------------------------------------------------------
// GEMM: C = A(4096x1024) @ W(1024x1024) + bias, optional scatter-store into
// the reference's scrambled head layout (B,H,S,HD).
// Block: 256 threads = 8 waves. Block tile 128x64, wave tile 32x32.
// Async double-buffered LDS staging: loads of chunk k+1 overlap WMMA on k.
// ---------------------------------------------------------------------------
#define GM 128
#define GN 64
#define GK 16
#define APAD 4   // A row stride 20 floats: 16B-aligned B128 dsts, 0-conflict
#define BPAD 4   // B row stride 68 floats: 16B-aligned B128 dsts

__global__ __launch_bounds__(256)
void gemm_kernel(const float* __restrict__ A, const float* __restrict__ W,
                 const float* __restrict__ bias, float* __restrict__ C,
                 int scatter)
{
  const int Nt = E_;        // 1024
  const int Kt = E_;        // 1024

  __shared__ float As[2][GM][GK + APAD];   // row-major [m][k]
  __shared__ float Bs[2][GK][GN + BPAD];   // row-major [k][n]

  const int tid  = threadIdx.x;
  const int lane = tid & 31;
  const int wid  = tid >> 5;      // 0..7
  const int hi   = lane >> 4;     // half-wave: K pairs (0,1) vs (2,3)
  const int lm   = lane & 15;

  const int m0 = blockIdx.y * GM;
  const int n0 = blockIdx.x * GN;
  const int wm = (wid >> 1) * 32; // wave M offset: 0,32,64,96
  const int wn = (wid & 1) * 32;  // wave N offset: 0,32

  v8f acc[4];
  v8f zro = {};
  acc[0] = zro; acc[1] = zro; acc[2] = zro; acc[3] = zro;

  // async staging assignment (per thread: 2x A-B128, 1x B-B128 per chunk)
  const int ar = tid >> 1;         // 0..127 : A row within tile
  const int ak = (tid & 1) * 8;    // 0/8    : A col group
  const int br = tid >> 4;         // 0..15  : B row within chunk
  const int bc = (tid & 15) * 4;   // B col group

  auto issue_tiles = [&](int buf, int k0) {
    const float* ga = A + (size_t)(m0 + ar) * Kt + k0 + ak;
    async_copy_b128(ga,     &As[buf][ar][ak]);
    async_copy_b128(ga + 4, &As[buf][ar][ak + 4]);
    const float* gb = W + (size_t)(k0 + br) * Nt + n0 + bc;
    async_copy_b128(gb, &Bs[buf][br][bc]);
  };

  issue_tiles(0, 0);
  const int nch = Kt / GK;   // 64
  for (int chunk = 0; chunk < nch; ++chunk) {
    const int buf = chunk & 1;
    WAIT_ASYNC(0x0);          // own async loads for `buf` arrived
    __syncthreads();          // everyone's arrived; prev compute done
    if (chunk + 1 < nch) issue_tiles(buf ^ 1, (chunk + 1) * GK);

#pragma unroll
    for (int ko = 0; ko < GK; ko += 4) {
      const int kk = ko + 2 * hi;
      v2f a0 = *(const v2f*)&As[buf][wm + 0  + lm][kk];
      v2f a1 = *(const v2f*)&As[buf][wm + 16 + lm][kk];
      v2f b0, b1;
      b0.x = Bs[buf][kk + 0][wn + 0  + lm];
      b0.y = Bs[buf][kk + 1][wn + 0  + lm];
      b1.x = Bs[buf][kk + 0][wn + 16 + lm];
      b1.y = Bs[buf][kk + 1][wn + 16 + lm];
      acc[0] = wmma_f32(a0, b0, acc[0]);
      acc[1] = wmma_f32(a0, b1, acc[1]);
      acc[2] = wmma_f32(a1, b0, acc[2]);
      acc[3] = wmma_f32(a1, b1, acc[3]);
    }
  }

#pragma unroll
  for (int f = 0; f < 4; ++f) {
    const int mo = (f >> 1) * 16;
    const int no = (f & 1) * 16;
#pragma unroll
    for (int r = 0; r < 8; ++r) {
      const int row = m0 + wm + mo + r + hi * 8;   // = b*S + s
      const int col = n0 + wn + no + lm;           // = e
      const float v = acc[f][r] + bias[col];
      if (!scatter) {
        C[(size_t)row * Nt + col] = v;
      } else {
        // faithful _create_heads permutation (transpose(2,1,0).reshape)
        const int b  = row >> 11;          // / S_
        const int s  = row & (S_ - 1);
        const int e  = col;
        const int b2 = e >> 9;
        const int h  = (e >> 5) & 15;
        const int t  = ((e & 31) << 12) + (s << 1) + b;
        const int s2 = t >> 6;
        const int d  = t & 63;
        C[((size_t)((b2 * H_ + h) * S_ + s2)) * HD_ + d] = v;
      }
    }
  }
}

// ---------------------------------------------------------------------------
// L2-normalize each HD=64 row of Qh and Kh in place. One wave per row.
// ---------------------------------------------------------------------------
__global__ __launch_bounds__(256)
void normalize_rows_kernel(float* __restrict__ Qh, float* __restrict__ Kh)
{
  const int lane = threadIdx.x & 31;
  const int wid  = threadIdx.x >> 5;
  const size_t row = (size_t)blockIdx.x * 8 + wid;   // 0 .. B*H*S-1
  float* P = (blockIdx.y == 0) ? Qh : Kh;
  float2* p = (float2*)(P + row * HD_) + lane;
  float2 x = *p;
  float ss = x.x * x.x + x.y * x.y;
#pragma unroll
  for (int off = 1; off < 32; off <<= 1) ss += __shfl_xor(ss, off, 32);
  const float r = rsqrtf(ss);
  x.x *= r; x.y *= r;
  *p = x;
}

// ---------------------------------------------------------------------------
// Pass 1 of softmax: per-row running max m and sum exp(l-m) over the causal
// range. One wave owns 16 query rows. K tiles are async-staged into per-wave
// double-buffered LDS (no block barriers: causal trip counts differ by wave).
// Row stats live across the 16 lanes sharing `hi`; merged via a log-sum-exp
// shfl_xor tree (NEG sentinel keeps exp() NaN-free when both sides empty).
// ---------------------------------------------------------------------------
#define KPAD 4   // K/V tile row stride 68 floats

__global__ __launch_bounds__(256)
void attn_stats_kernel(const float* __restrict__ Qh, const float* __restrict__ Kh,
                       float* __restrict__ Mrow, float* __restrict__ Srow)
{
  __shared__ float Kt[8][2][16][HD_ + KPAD];

  const int lane  = threadIdx.x & 31;
  const int wid   = threadIdx.x >> 5;
  const int hi    = lane >> 4;
  const int lm    = lane & 15;
  const int bh    = blockIdx.y;
  const int itile = blockIdx.x * 8 + wid;
  const int ibase = itile * 16;

  const float* Q = Qh + (size_t)bh * S_ * HD_;
  const float* K = Kh + (size_t)bh * S_ * HD_;

  v2f qa[16];
#pragma unroll
  for (int ko = 0; ko < 16; ++ko) {
    qa[ko].x = Q[(size_t)(ibase + lm) * HD_ + ko * 4 + 2 * hi + 0];
    qa[ko].y = Q[(size_t)(ibase + lm) * HD_ + ko * 4 + 2 * hi + 1];
  }

  auto issueK = [&](int buf, int jbase) {     // 8 B128 per lane = 4KB tile
#pragma unroll
    for (int t = 0; t < 8; ++t) {
      const int flat = t * 32 + lane;
      const int row  = flat >> 4;             // 0..15
      const int c4   = (flat & 15) * 4;
      async_copy_b128(K + (size_t)(jbase + row) * HD_ + c4,
                      &Kt[wid][buf][row][c4]);
    }
  };

  const float NEG = -3.0e38f;
  float rm[8], rs[8];
#pragma unroll
  for (int r = 0; r < 8; ++r) { rm[r] = NEG; rs[r] = 0.f; }

  issueK(0, 0);
  for (int jt = 0; jt <= itile; ++jt) {
    const int buf = jt & 1;
    if (jt < itile) { issueK(buf ^ 1, (jt + 1) * 16); WAIT_ASYNC(0x8); }
    else           { WAIT_ASYNC(0x0); }

    v8f c = {};
#pragma unroll
    for (int ko = 0; ko < 16; ++ko) {
      v2f b = *(const v2f*)&Kt[wid][buf][lm][ko * 4 + 2 * hi];
      c = wmma_f32(qa[ko], b, c);
    }
    const bool diag = (jt == itile);
#pragma unroll
    for (int r = 0; r < 8; ++r) {
      const float l = c[r] * 1000.f;
      const bool valid = !diag || (lm <= r + hi * 8);   // causal
      if (valid) {
        if (l > rm[r]) { rs[r] = rs[r] * __expf(rm[r] - l) + 1.f; rm[r] = l; }
        else           { rs[r] += __expf(l - rm[r]); }
      }
    }
  }

#pragma unroll
  for (int off = 1; off < 16; off <<= 1) {
#pragma unroll
    for (int r = 0; r < 8; ++r) {
      const float om = __shfl_xor(rm[r], off, 32);
      const float os = __shfl_xor(rs[r], off, 32);
      const float nm = fmaxf(rm[r], om);
      rs[r] = rs[r] * __expf(rm[r] - nm) + os * __expf(om - nm);
      rm[r] = nm;
    }
  }

  if (lm == 0) {
#pragma unroll
    for (int r = 0; r < 8; ++r) {
      const int row = ibase + r + hi * 8;
      Mrow[(size_t)bh * S_ + row] = rm[r];
      Srow[(size_t)bh * S_ + row] = rs[r];
    }
  }
}

// ---------------------------------------------------------------------------
// Pass 2: recompute logit tiles, write normalized attn (dominant HBM write),
// accumulate z = p @ V with WMMA. K and V tiles async-staged per wave
// (double-buffered); p-tile bounced through per-wave LDS to convert
// C-layout -> A-layout. z stored directly in zc (B,S,E) layout.
// ---------------------------------------------------------------------------
__global__ __launch_bounds__(256)
void attn_out_kernel(const float* __restrict__ Qh, const float* __restrict__ Kh,
                     const float* __restrict__ Vh,
                     const float* __restrict__ Mrow, const float* __restrict__ Srow,
                     float* __restrict__ attnOut, float* __restrict__ Zc)
{
  __shared__ float Kt[8][2][16][HD_ + KPAD];
  __shared__ float Vt[8][2][16][HD_ + KPAD];
  __shared__ float pS[8][16][17];

  const int lane  = threadIdx.x & 31;
  const int wid   = threadIdx.x >> 5;
  const int hi    = lane >> 4;
  const int lm    = lane & 15;
  const int bh    = blockIdx.y;
  const int b2    = bh >> 4;
  const int h     = bh & 15;
  const int itile = blockIdx.x * 8 + wid;
  const int ibase = itile * 16;

  const float* Q = Qh + (size_t)bh * S_ * HD_;
  const float* K = Kh + (size_t)bh * S_ * HD_;
  const float* V = Vh + (size_t)bh * S_ * HD_;
  float* aout = attnOut + (size_t)bh * S_ * S_;

  v2f qa[16];
#pragma unroll
  for (int ko = 0; ko < 16; ++ko) {
    qa[ko].x = Q[(size_t)(ibase + lm) * HD_ + ko * 4 + 2 * hi + 0];
    qa[ko].y = Q[(size_t)(ibase + lm) * HD_ + ko * 4 + 2 * hi + 1];
  }

  float rm[8], rinv[8];
#pragma unroll
  for (int r = 0; r < 8; ++r) {
    const int row = ibase + r + hi * 8;
    rm[r]   = Mrow[(size_t)bh * S_ + row];
    rinv[r] = 1.0f / Srow[(size_t)bh * S_ + row];
  }

  auto issueKV = [&](int buf, int jbase) {    // 16 B128 per lane (K + V)
#pragma unroll
    for (int t = 0; t < 8; ++t) {
      const int flat = t * 32 + lane;
      const int row  = flat >> 4;
      const int c4   = (flat & 15) * 4;
      async_copy_b128(K + (size_t)(jbase + row) * HD_ + c4,
                      &Kt[wid][buf][row][c4]);
      async_copy_b128(V + (size_t)(jbase + row) * HD_ + c4,
                      &Vt[wid][buf][row][c4]);
    }
  };

  v8f zacc[4];
  v8f zro = {};
  zacc[0] = zro; zacc[1] = zro; zacc[2] = zro; zacc[3] = zro;

  issueKV(0, 0);
  for (int jt = 0; jt < S_ / 16; ++jt) {
    const int jbase = jt * 16;
    if (jt <= itile) {               // wave-uniform branch (EXEC stays full)
      const int buf = jt & 1;
      if (jt < itile) { issueKV(buf ^ 1, jbase + 16); WAIT_ASYNC(0x10); }
      else            { WAIT_ASYNC(0x0); }

      v8f c = {};
#pragma unroll
      for (int ko = 0; ko < 16; ++ko) {
        v2f b = *(const v2f*)&Kt[wid][buf][lm][ko * 4 + 2 * hi];
        c = wmma_f32(qa[ko], b, c);
      }
#pragma unroll
      for (int r = 0; r < 8; ++r) {
        const int row = ibase + r + hi * 8;
        const int col = jbase + lm;
        const float l  = c[r] * 1000.f;
        const float pv = (col <= row) ? __expf(l - rm[r]) * rinv[r] : 0.f;
        aout[(size_t)row * S_ + col] = pv;   // coalesced 64B per half-wave
        pS[wid][r + hi * 8][lm] = pv;        // C-layout -> LDS
      }
      // z += p(16x16) @ V(16x64) : 4 K-steps x 4 HD chunks
#pragma unroll
      for (int kk = 0; kk < 16; kk += 4) {
        v2f pa = *(const v2f*)&pS[wid][lm][kk + 2 * hi];   // A-layout read
#pragma unroll
        for (int nc = 0; nc < 4; ++nc) {
          v2f vb;
          vb.x = Vt[wid][buf][kk + 2 * hi + 0][nc * 16 + lm];
          vb.y = Vt[wid][buf][kk + 2 * hi + 1][nc * 16 + lm];
          zacc[nc] = wmma_f32(pa, vb, zacc[nc]);
        }
      }
    } else {
      // strictly-masked tiles: attn is exactly zero
#pragma unroll
      for (int r = 0; r < 8; ++r)
        aout[(size_t)(ibase + r + hi * 8) * S_ + jbase + lm] = 0.f;
    }
  }

  // zc[b2][row][h*64 + d] = z[b2][h][row][d]  (transpose fused into store)
#pragma unroll
  for (int nc = 0; nc < 4; ++nc) {
#pragma unroll
    for (int r = 0; r < 8; ++r) {
      const int row = ibase + r + hi * 8;
      const int d   = nc * 16 + lm;
      Zc[((size_t)(b2 * S_ + row)) * E_ + h * HD_ + d] = zacc[nc][r];
    }
  }
}

// ---------------------------------------------------------------------------
extern "C" void kernel_launch(void* const* d_in, const int* in_sizes, int n_in,
                              void* d_out, int out_size, void* d_ws, size_t ws_size,
                              hipStream_t stream)
{
  const float* q  = (const float*)d_in[0];
  const float* k  = (const float*)d_in[1];
  const float* v  = (const float*)d_in[2];
  // d_in[3] = mask : causal structure handled analytically
  const float* Wq = (const float*)d_in[4];
  const float* bq = (const float*)d_in[5];
  const float* Wk = (const float*)d_in[6];
  const float* bk = (const float*)d_in[7];
  const float* Wv = (const float*)d_in[8];
  const float* bv = (const float*)d_in[9];
  const float* Wz = (const float*)d_in[10];
  const float* bz = (const float*)d_in[11];

  float* out  = (float*)d_out;                      // (B,S,E)
  float* attn = out + (size_t)B_ * S_ * E_;         // (B,H,S,S)

  float* ws   = (float*)d_ws;
  float* Qh   = ws;                                 // (B,H,S,HD)
  float* Kh   = Qh + BSE_;
  float* Vh   = Kh + BSE_;
  float* Zc   = Vh + BSE_;                          // (B,S,E)
  float* Mrow = Zc + BSE_;                          // (B*H*S)
  float* Srow = Mrow + (size_t)BH_ * S_;

  const dim3 blk(256);
  const dim3 ggrid(E_ / GN, (B_ * S_) / GM);        // 16 x 32

  gemm_kernel<<<ggrid, blk, 0, stream>>>(q, Wq, bq, Qh, 1);
  gemm_kernel<<<ggrid, blk, 0, stream>>>(k, Wk, bk, Kh, 1);
  gemm_kernel<<<ggrid, blk, 0, stream>>>(v, Wv, bv, Vh, 1);

  normalize_rows_kernel<<<dim3(BH_ * S_ / 8, 2), blk, 0, stream>>>(Qh, Kh);

  attn_stats_kernel<<<dim3(S_ / 128, BH_), blk, 0, stream>>>(Qh, Kh, Mrow, Srow);
  attn_out_kernel  <<<dim3(S_ / 128, BH_), blk, 0, stream>>>(Qh, Kh, Vh, Mrow, Srow,
                                                             attn, Zc);

  gemm_kernel<<<ggrid, blk, 0, stream>>>(Zc, Wz, bz, out, 0);
}